// SSIMLoss_58634893525441
// MI455X (gfx1250) — compile-verified
//
#include <hip/hip_runtime.h>

typedef __attribute__((ext_vector_type(2))) float v2f;
typedef __attribute__((ext_vector_type(4))) float v4f;
typedef __attribute__((ext_vector_type(8))) float v8f;

#define BATCH 16
#define CHANS 3
#define IMH 512
#define IMW 512
#define TILE 16
#define XROWS 32          // 26 valid rows (16 out + 10 halo), zero-padded to 32
#define XCOLS 32          // 26 valid cols, zero-padded to 32 (power-of-2 addressing)
#define TCOLS 32          // transposed-T row stride
#define NTILES ((BATCH*CHANS) * (IMH/TILE) * (IMW/TILE))   // 49152
#define WAVES_PER_BLOCK 4
#define LDS_PER_WAVE (2*XROWS*XCOLS + TILE*TCOLS)          // 2560 floats = 10240 B

// sum of the 11 unnormalized Gaussian taps, sigma = 1.5
#define GSUM 3.7592338f

__global__ __launch_bounds__(32 * WAVES_PER_BLOCK)
void ssim_tiles_wmma(const float* __restrict__ pred,
                     const float* __restrict__ label,
                     float* __restrict__ acc)
{
    __shared__ float smem[WAVES_PER_BLOCK * LDS_PER_WAVE];

    const int lane = threadIdx.x & 31;
    const int wave = threadIdx.x >> 5;
    const int m    = lane & 15;   // M/N index within lane-half (ISA WMMA layout)
    const int hi   = lane >> 4;   // lane-half: selects K offset (+2)

    float* xs = &smem[wave * LDS_PER_WAVE];
    float* ys = xs + XROWS * XCOLS;
    float* ts = ys + XROWS * XCOLS;        // T stored transposed: ts[col*32 + row]

    const int tile   = blockIdx.x * WAVES_PER_BLOCK + wave;
    const int bc     = tile >> 10;                 // / (32*32 tiles per channel)
    const int within = tile & 1023;
    const int h0     = (within >> 5) << 4;
    const int w0     = (within & 31) << 4;
    const float* pbase = pred  + (size_t)bc * (IMH * IMW);
    const float* lbase = label + (size_t)bc * (IMH * IMW);

    // speculative prefetch of tile start (global_prefetch_b8)
    __builtin_prefetch(pbase + h0 * IMW + w0, 0, 3);

    // ---- stage zero-padded 26x26 patches into wave-private LDS (32x32) ----
    // row r = iter, col c = lane: pure shift/mask addressing, no int division
    for (int r = 0; r < XROWS; ++r) {
        int gh = h0 - 5 + r;
        int gw = w0 - 5 + lane;
        float xv = 0.0f, yv = 0.0f;
        if (r < 26 && lane < 26 && gh >= 0 && gh < IMH && gw >= 0 && gw < IMW) {
            xv = pbase[gh * IMW + gw];
            yv = lbase[gh * IMW + gw];
        }
        xs[r * XCOLS + lane] = xv;
        ys[r * XCOLS + lane] = yv;
    }
    // LDS region is wave-private and same-wave DS ops are in order: no barrier.

    // ---- normalized Gaussian taps distributed across lanes ----
    // lane l holds g[l] for l in [0,10], zero otherwise. One expf per lane.
    float gv = 0.0f;
    if (lane < 11) {
        float f = (float)(lane - 5);
        gv = __expf(-f * f * (1.0f / 4.5f)) * (1.0f / GSUM);
    }

    // Banded-matrix fragments, identical for both stages:
    //   horizontal B operand: H[k,j] = g[k-j], j = m
    //   vertical   A operand: G[i,k] = g[k-i], i = m
    // component = g[d] with d = 4g+2*hi+{0,1} - m; d in [-15,27] wraps to a
    // zero-holding lane for every out-of-band offset -> single shuffle each.
    v2f wfrag[7];
#pragma unroll
    for (int g = 0; g < 7; ++g) {
        int k = 4 * g + 2 * hi;
        wfrag[g].x = __shfl(gv, (k     - m) & 31, 32);
        wfrag[g].y = __shfl(gv, (k + 1 - m) & 31, 32);
    }

    // ---- hoisted A-operand loads (shared by all 5 conv quantities) ----
    v2f xtv[7], ytv[7], xbv[7], ybv[7];
#pragma unroll
    for (int g = 0; g < 7; ++g) {
        int k = 4 * g + 2 * hi;
        xtv[g] = *(const v2f*)&xs[m * XCOLS + k];          // A: lane m = row
        ytv[g] = *(const v2f*)&ys[m * XCOLS + k];
        xbv[g] = *(const v2f*)&xs[(16 + m) * XCOLS + k];
        ybv[g] = *(const v2f*)&ys[(16 + m) * XCOLS + k];
    }

    // ---- 5 depthwise convs as separable WMMA pipelines ----
    v8f O[5];
#pragma unroll
    for (int q = 0; q < 5; ++q) {
        v8f Ttop = {};
        v8f Tbot = {};
        // Horizontal: T = X * H    (two 16-row blocks, K = 28 in 7 steps)
#pragma unroll
        for (int g = 0; g < 7; ++g) {
            v2f at, ab;
            if      (q == 0) { at = xtv[g];          ab = xbv[g];          }
            else if (q == 1) { at = ytv[g];          ab = ybv[g];          }
            else if (q == 2) { at = xtv[g] * xtv[g]; ab = xbv[g] * xbv[g]; }
            else if (q == 3) { at = ytv[g] * ytv[g]; ab = ybv[g] * ybv[g]; }
            else             { at = xtv[g] * ytv[g]; ab = xbv[g] * ybv[g]; }
            Ttop = __builtin_amdgcn_wmma_f32_16x16x4_f32(
                       false, at, false, wfrag[g], (short)0, Ttop, false, false);
            Tbot = __builtin_amdgcn_wmma_f32_16x16x4_f32(
                       false, ab, false, wfrag[g], (short)0, Tbot, false, false);
        }
        // C/D layout -> B layout via transposed LDS store (b128 vectorized):
        // D elem (M = p + 8*hi, N = m) in VGPR p; write ts[N*32 + M]
        {
            v4f t0 = { Ttop[0], Ttop[1], Ttop[2], Ttop[3] };
            v4f t1 = { Ttop[4], Ttop[5], Ttop[6], Ttop[7] };
            v4f b0 = { Tbot[0], Tbot[1], Tbot[2], Tbot[3] };
            v4f b1 = { Tbot[4], Tbot[5], Tbot[6], Tbot[7] };
            float* trow = &ts[m * TCOLS + 8 * hi];
            *(v4f*)(trow)          = t0;   // rows  8*hi   .. 8*hi+3
            *(v4f*)(trow + 4)      = t1;   // rows  8*hi+4 .. 8*hi+7
            *(v4f*)(trow + 16)     = b0;   // rows 16+8*hi ..
            *(v4f*)(trow + 20)     = b1;
        }
        // Vertical: O = G * T. K-group 7 (K=28..31) hits all-zero G columns
        // and all-zero T rows -> skipped. 7 groups cover K = 0..27.
        v8f o = {};
#pragma unroll
        for (int g = 0; g < 7; ++g) {
            int k = 4 * g + 2 * hi;
            v2f bt = *(const v2f*)&ts[m * TCOLS + k];
            o = __builtin_amdgcn_wmma_f32_16x16x4_f32(
                    false, wfrag[g], false, bt, (short)0, o, false, false);
        }
        O[q] = o;
    }

    // ---- SSIM map on aligned D fragments, then reduce ----
    const float C1 = 0.0001f;   // 0.01^2
    const float C2 = 0.0009f;   // 0.03^2
    float ssum = 0.0f;
#pragma unroll
    for (int e = 0; e < 8; ++e) {
        float mu1  = O[0][e];
        float mu2  = O[1][e];
        float mu1s = mu1 * mu1;
        float mu2s = mu2 * mu2;
        float mu12 = mu1 * mu2;
        float s1   = O[2][e] - mu1s;
        float s2   = O[3][e] - mu2s;
        float s12  = O[4][e] - mu12;
        float num  = (2.0f * mu12 + C1) * (2.0f * s12 + C2);
        float den  = (mu1s + mu2s + C1) * (s1 + s2 + C2);
        ssum += num / den;
    }
    // wave32 butterfly reduction
#pragma unroll
    for (int off = 16; off > 0; off >>= 1)
        ssum += __shfl_xor(ssum, off, 32);
    if (lane == 0)
        atomicAdd(acc, ssum);
}

__global__ void ssim_zero_acc(float* acc) {
    if (threadIdx.x == 0 && blockIdx.x == 0) acc[0] = 0.0f;
}

__global__ void ssim_finalize(const float* __restrict__ acc, float* __restrict__ out) {
    if (threadIdx.x == 0 && blockIdx.x == 0) {
        const float invN = 1.0f / ((float)BATCH * CHANS * IMH * IMW);
        out[0] = 1.0f - acc[0] * invN;
    }
}

extern "C" void kernel_launch(void* const* d_in, const int* in_sizes, int n_in,
                              void* d_out, int out_size, void* d_ws, size_t ws_size,
                              hipStream_t stream) {
    const float* pred  = (const float*)d_in[0];
    const float* label = (const float*)d_in[1];
    float* acc = (float*)d_ws;

    ssim_zero_acc<<<1, 32, 0, stream>>>(acc);

    const int nblocks = NTILES / WAVES_PER_BLOCK;   // 12288 blocks x 128 threads
    ssim_tiles_wmma<<<nblocks, 32 * WAVES_PER_BLOCK, 0, stream>>>(pred, label, acc);

    ssim_finalize<<<1, 32, 0, stream>>>(acc, (float*)d_out);
}